// InterpolationNetwork_11922829213820
// MI455X (gfx1250) — compile-verified
//
#include <hip/hip_runtime.h>
#include <hip/hip_bf16.h>

// TAGConv (K=50) x 4 layers on MI455X (gfx1250, wave32).
// Sparse hop: edge-parallel fp32 atomics, b128 gathers; entire graph state
// (<32MB) is L2-resident (192MB L2), so gathers+atomics resolve in L2.
// Dense projection: fully-unrolled V_WMMA_F32_16X16X4_F32 chains, A via
// unconditional global b64 loads, B via transposed zero-padded W tile in LDS.
// Per-hop buffer clears and bias init are fused into the GEMM kernels to cut
// graph launch count (~413 kernels) and N*SO roundtrips.

#define NN 100000
#define EE 1600000
#define KHOPS 50
#define MTILES (NN / 16)   // 6250 exact

typedef __attribute__((ext_vector_type(2))) float v2f;
typedef __attribute__((ext_vector_type(8))) float v8f;

__host__ __device__ constexpr int up4(int x)  { return (x + 3) & ~3; }
__host__ __device__ constexpr int up16(int x) { return (x + 15) & ~15; }

__global__ void zero_kernel(float* __restrict__ p, int n) {
    int t = blockIdx.x * blockDim.x + threadIdx.x;
    if (t < n) p[t] = 0.0f;
}

__global__ void deg_kernel(const int* __restrict__ col, const float* __restrict__ w,
                           float* __restrict__ deg) {
    int e = blockIdx.x * blockDim.x + threadIdx.x;
    if (e < EE) atomicAdd(&deg[col[e]], w[e]);
}

__global__ void dis_kernel(float* __restrict__ deg) {
    int n = blockIdx.x * blockDim.x + threadIdx.x;
    if (n < NN) {
        float d = deg[n];
        deg[n] = (d > 0.0f) ? rsqrtf(d) : 0.0f;   // in-place deg -> D^{-1/2}
    }
}

__global__ void norm_kernel(const int* __restrict__ row, const int* __restrict__ col,
                            const float* __restrict__ w, const float* __restrict__ dis,
                            float* __restrict__ nrm) {
    int e = blockIdx.x * blockDim.x + threadIdx.x;
    if (e < EE) nrm[e] = dis[row[e]] * w[e] * dis[col[e]];
}

// x = [one_hot(i), c], stored with stride 16, pad columns = 0.
__global__ void xinit_kernel(const float* __restrict__ c, const int* __restrict__ iidx,
                             float* __restrict__ x) {
    int n = blockIdx.x * blockDim.x + threadIdx.x;
    if (n < NN) {
        float4* p = (float4*)(x + (size_t)n * 16);
        p[0] = make_float4((n == iidx[0]) ? 1.0f : 0.0f, c[n], 0.0f, 0.0f);
        p[1] = make_float4(0.f, 0.f, 0.f, 0.f);
        p[2] = make_float4(0.f, 0.f, 0.f, 0.f);
        p[3] = make_float4(0.f, 0.f, 0.f, 0.f);
    }
}

// One thread per edge. Vectorized b128 gather of the (padded) source row,
// scalar non-returning fp32 atomics into the real columns of the dest row.
template <int FIN, int S>
__global__ void spmm_kernel(const int* __restrict__ row, const int* __restrict__ col,
                            const float* __restrict__ nrm, const float* __restrict__ h,
                            float* __restrict__ hn) {
    int e = blockIdx.x * blockDim.x + threadIdx.x;
    if (e >= EE) return;
    const int r = row[e];
    const int d = col[e];
    const float wv = nrm[e];

    constexpr int NV = (FIN + 3) / 4;      // float4 chunks covering real cols
    float4 v[NV];
    const float4* __restrict__ hr = (const float4*)(h + (size_t)r * S);
#pragma unroll
    for (int q = 0; q < NV; ++q) v[q] = hr[q];

    float* __restrict__ ho = hn + (size_t)d * S;
    const float* vs = (const float*)v;
#pragma unroll
    for (int f = 0; f < FIN; ++f)
        atomicAdd(&ho[f], wv * vs[f]);
}

// out[NN,FOUT(pad SO)] (+)= H[NN,FIN(pad SH)] @ W[FIN,FOUT]
// INIT: seed C fragment with bias (pads 0) instead of loading `out`.
// One wave per 16x16 tile; W slice staged transposed + zero-padded in LDS.
// A frag: lanes 0-15 -> K=kk,kk+1 of row m0+lane; lanes 16-31 -> K=kk+2,kk+3.
// C/D frag: VGPR r -> row m0 + r + 8*(lane>=16), col n0 + (lane&15).
// Tail: grid-stride b128 clear of `zbuf` (dead ping-pong buffer for next hop).
template <int FIN, int FOUT, bool INIT>
__global__ __launch_bounds__(128) void gemm_acc_wmma(const float* __restrict__ H,
                                                     const float* __restrict__ W,
                                                     const float* __restrict__ bias,
                                                     float* __restrict__ out,
                                                     float4* __restrict__ zbuf,
                                                     int zn4) {
    constexpr int FINP = up4(FIN);
    constexpr int SH = up16(FIN);
    constexpr int SO = up16(FOUT);
    __shared__ float Wt[SO][FINP];         // Wt[j][k] = W[k][j], zero-padded

    for (int t = threadIdx.x; t < SO * FINP; t += 128) {
        int j = t / FINP, k = t % FINP;
        Wt[j][k] = (j < FOUT && k < FIN) ? W[k * FOUT + j] : 0.0f;
    }
    __syncthreads();

    const int lane = threadIdx.x & 31;
    const int mtile = blockIdx.x * 4 + (threadIdx.x >> 5);
    if (mtile < MTILES) {                  // wave-uniform: EXEC all-ones inside
        const int m0 = mtile * 16;
        const int n0 = blockIdx.y * 16;
        const int half = lane >> 4;
        const int l16 = lane & 15;
        const int ccol = n0 + l16;         // always < SO

        v8f c;
        if constexpr (INIT) {
            const float bv = (ccol < FOUT) ? bias[ccol] : 0.0f;
#pragma unroll
            for (int r = 0; r < 8; ++r) c[r] = bv;
        } else {
#pragma unroll
            for (int r = 0; r < 8; ++r)
                c[r] = out[(size_t)(m0 + r + half * 8) * SO + ccol];
        }

        const float* __restrict__ hrow = H + (size_t)(m0 + l16) * SH;
#pragma unroll
        for (int kk = 0; kk < FINP; kk += 4) {
            const int k0 = kk + half * 2;
            v2f a = *(const v2f*)(hrow + k0);          // global_load_b64
            v2f b = *(const v2f*)(&Wt[ccol][k0]);      // ds_load_b64
            // 8 args: (neg_a, A, neg_b, B, c_mod, C, reuse_a, reuse_b)
            c = __builtin_amdgcn_wmma_f32_16x16x4_f32(false, a, false, b, (short)0,
                                                      c, false, false);
        }

#pragma unroll
        for (int r = 0; r < 8; ++r)
            out[(size_t)(m0 + r + half * 8) * SO + ccol] = c[r];
    }

    // Fused clear of the scatter target for the NEXT hop (b128 stores).
    const int gtid = (blockIdx.y * gridDim.x + blockIdx.x) * 128 + threadIdx.x;
    const int gstride = gridDim.x * gridDim.y * 128;
    const float4 z4 = make_float4(0.f, 0.f, 0.f, 0.f);
    for (int t = gtid; t < zn4; t += gstride) zbuf[t] = z4;
}

__global__ void relu_kernel(const float* __restrict__ src, float* __restrict__ dst,
                            int n) {
    int t = blockIdx.x * blockDim.x + threadIdx.x;
    if (t < n) dst[t] = fmaxf(src[t], 0.0f);
}

// Final layer: compact d_out[n] = relu(out[n*16 + 0])
__global__ void relu_extract_kernel(const float* __restrict__ src,
                                    float* __restrict__ dst) {
    int n = blockIdx.x * blockDim.x + threadIdx.x;
    if (n < NN) dst[n] = fmaxf(src[(size_t)n * 16], 0.0f);
}

template <int FIN, int FOUT>
static void run_layer(const int* row, const int* colp, const float* nrm,
                      const float* W, const float* b,
                      float* bufA, float* bufB, float* bufO, hipStream_t stream) {
    constexpr int SH = up16(FIN);
    constexpr int SO = up16(FOUT);
    const int TB = 256;
    const int gE = (EE + TB - 1) / TB;
    dim3 ggrid((MTILES + 3) / 4, SO / 16);
    constexpr int ZN4 = NN * SH / 4;

    // k=0: out = bias + x @ W[0]; also clears bufB (hop-1 scatter target).
    gemm_acc_wmma<FIN, FOUT, true><<<ggrid, 128, 0, stream>>>(
        bufA, W, b, bufO, (float4*)bufB, ZN4);

    float* cur = bufA;
    float* nxt = bufB;
    for (int k = 1; k <= KHOPS; ++k) {
        spmm_kernel<FIN, SH><<<gE, TB, 0, stream>>>(row, colp, nrm, cur, nxt);
        // `cur` is dead after the spmm above; clear it for hop k+1.
        const int zn4 = (k < KHOPS) ? ZN4 : 0;
        gemm_acc_wmma<FIN, FOUT, false><<<ggrid, 128, 0, stream>>>(
            nxt, W + (size_t)k * FIN * FOUT, b, bufO, (float4*)cur, zn4);
        float* t = cur; cur = nxt; nxt = t;
    }
}

extern "C" void kernel_launch(void* const* d_in, const int* in_sizes, int n_in,
                              void* d_out, int out_size, void* d_ws, size_t ws_size,
                              hipStream_t stream) {
    const int* row = (const int*)d_in[0];          // edge_index[0]
    const int* colp = row + EE;                    // edge_index[1]
    const float* ew = (const float*)d_in[1];
    const float* c = (const float*)d_in[2];
    const int* ii = (const int*)d_in[3];
    const float* W1 = (const float*)d_in[4];
    const float* b1 = (const float*)d_in[5];
    const float* W2 = (const float*)d_in[6];
    const float* b2 = (const float*)d_in[7];
    const float* W3 = (const float*)d_in[8];
    const float* b3 = (const float*)d_in[9];
    const float* W4 = (const float*)d_in[10];
    const float* b4 = (const float*)d_in[11];

    // Workspace (floats): deg[N] | norm[E] | bufA[N*32] | bufB[N*32] | bufO[N*32]
    float* ws = (float*)d_ws;
    float* deg = ws;
    float* nrm = ws + NN;
    float* bufA = nrm + EE;
    float* bufB = bufA + (size_t)NN * 32;
    float* bufO = bufB + (size_t)NN * 32;

    const int TB = 256;
    const int gN = (NN + TB - 1) / TB;
    const int gE = (EE + TB - 1) / TB;

    // gcn_norm
    zero_kernel<<<gN, TB, 0, stream>>>(deg, NN);
    deg_kernel<<<gE, TB, 0, stream>>>(colp, ew, deg);
    dis_kernel<<<gN, TB, 0, stream>>>(deg);
    norm_kernel<<<gE, TB, 0, stream>>>(row, colp, ew, deg, nrm);

    // x = [one_hot(i), c] with stride 16
    xinit_kernel<<<gN, TB, 0, stream>>>(c, ii, bufA);

    // Layer 1: 2 -> 15
    run_layer<2, 15>(row, colp, nrm, W1, b1, bufA, bufB, bufO, stream);
    relu_kernel<<<(NN * 16 + TB - 1) / TB, TB, 0, stream>>>(bufO, bufA, NN * 16);

    // Layer 2: 15 -> 30
    run_layer<15, 30>(row, colp, nrm, W2, b2, bufA, bufB, bufO, stream);
    relu_kernel<<<(NN * 32 + TB - 1) / TB, TB, 0, stream>>>(bufO, bufA, NN * 32);

    // Layer 3: 30 -> 15
    run_layer<30, 15>(row, colp, nrm, W3, b3, bufA, bufB, bufO, stream);
    relu_kernel<<<(NN * 16 + TB - 1) / TB, TB, 0, stream>>>(bufO, bufA, NN * 16);

    // Layer 4: 15 -> 1, compact output
    run_layer<15, 1>(row, colp, nrm, W4, b4, bufA, bufB, bufO, stream);
    relu_extract_kernel<<<gN, TB, 0, stream>>>(bufO, (float*)d_out);
}